// VolumeNormalizer_49555332662150
// MI455X (gfx1250) — compile-verified
//
#include <hip/hip_runtime.h>
#include <hip/hip_bf16.h>
#include <stdint.h>

#define NBATCH 64
#define TPB 256
#define TRI_PER_BLOCK 1024

#if defined(__has_builtin)
#if __has_builtin(__builtin_amdgcn_global_load_async_to_lds_b64) && \
    __has_builtin(__builtin_amdgcn_s_wait_asynccnt)
#define HAVE_ASYNC_LDS 1
#endif
#if __has_builtin(__builtin_amdgcn_wmma_f32_16x16x4_f32)
#define HAVE_WMMA_F32X4 1
#endif
#endif

typedef float v2f __attribute__((ext_vector_type(2)));
typedef float v8f __attribute__((ext_vector_type(8)));
// Exact pointee type the async-LDS builtin wants (per compiler diagnostic):
typedef int v2i_a __attribute__((vector_size(2 * sizeof(int))));
typedef __attribute__((address_space(1))) v2i_a* gptr_v2i;
typedef __attribute__((address_space(3))) v2i_a* lptr_v2i;

// ---------------------------------------------------------------------------
// Kernel 1: per-(batch, chunk) partial sum of |det(v0, v1, v2)|.
// Indices for the chunk are staged into LDS with async global->LDS copies
// (ASYNCcnt path); vertex gathers hit L2 (entire x tensor fits in 192MB L2).
// ---------------------------------------------------------------------------
__global__ void __launch_bounds__(TPB) vol_partial_kernel(
    const float* __restrict__ x, const long long* __restrict__ M,
    float* __restrict__ partials, int nTri, int nChunks, int perBatch)
{
    __shared__ long long sidx[TRI_PER_BLOCK * 3];
    __shared__ float sred[TPB / 32];

    const int chunk = blockIdx.x;
    const int b = blockIdx.y;
    const long long tri0 = (long long)chunk * TRI_PER_BLOCK;
    const long long totIdx = (long long)nTri * 3;

#ifdef HAVE_ASYNC_LDS
    for (int i = threadIdx.x; i < TRI_PER_BLOCK * 3; i += TPB) {
        long long g = tri0 * 3 + i;
        if (g >= totIdx) g = totIdx - 1;   // clamp: stays in-bounds, value unused
        __builtin_amdgcn_global_load_async_to_lds_b64(
            (gptr_v2i)(M + g),
            (lptr_v2i)&sidx[i],
            0, 0);
    }
    __builtin_amdgcn_s_wait_asynccnt(0);
#else
    for (int i = threadIdx.x; i < TRI_PER_BLOCK * 3; i += TPB) {
        long long g = tri0 * 3 + i;
        if (g >= totIdx) g = totIdx - 1;
        sidx[i] = M[g];
    }
#endif
    __syncthreads();

    const float* __restrict__ xb = x + (size_t)b * perBatch;
    float acc = 0.0f;
#pragma unroll
    for (int k = 0; k < TRI_PER_BLOCK / TPB; ++k) {
        const int t = k * TPB + threadIdx.x;
        const long long gt = tri0 + t;
        if (gt < nTri) {
            const int i0 = (int)sidx[3 * t + 0];
            const int i1 = (int)sidx[3 * t + 1];
            const int i2 = (int)sidx[3 * t + 2];
            const float* p0 = xb + 3 * i0;
            const float* p1 = xb + 3 * i1;
            const float* p2 = xb + 3 * i2;
            const float ax = p0[0], ay = p0[1], az = p0[2];
            const float bx = p1[0], by = p1[1], bz = p1[2];
            const float cx = p2[0], cy = p2[1], cz = p2[2];
            // cross(v1, v2)
            const float rx = by * cz - bz * cy;
            const float ry = bz * cx - bx * cz;
            const float rz = bx * cy - by * cx;
            const float det = ax * rx + ay * ry + az * rz;
            acc += fabsf(det);
        }
    }

    // wave32 reduction, then cross-wave via LDS (deterministic order)
    for (int off = 16; off > 0; off >>= 1)
        acc += __shfl_xor(acc, off, 32);
    const int lane = threadIdx.x & 31;
    const int wid = threadIdx.x >> 5;
    if (lane == 0) sred[wid] = acc;
    __syncthreads();
    if (threadIdx.x == 0) {
        float s = 0.0f;
#pragma unroll
        for (int i = 0; i < TPB / 32; ++i) s += sred[i];
        partials[(size_t)b * nChunks + chunk] = s;
    }
}

// ---------------------------------------------------------------------------
// Kernel 2: one wave per batch. Strided lane accumulation of chunk partials,
// then a single v_wmma_f32_16x16x4_f32 against an all-ones B matrix performs
// the cross-lane sum: D[m][n] = sum_k A[m][k], so (sum of a lane's 8 C regs)
// + shfl_xor(...,16) == total over all 64 lane accumulators. EXEC is all 1s.
// ---------------------------------------------------------------------------
__global__ void __launch_bounds__(32) vol_finalize_kernel(
    const float* __restrict__ partials, float* __restrict__ invScale, int nChunks)
{
    const int b = blockIdx.x;
    const int lane = threadIdx.x;
    const float* __restrict__ p = partials + (size_t)b * nChunks;

    float a0 = 0.0f, a1 = 0.0f;
    for (int i = lane; i < nChunks; i += 64) a0 += p[i];
    for (int i = lane + 32; i < nChunks; i += 64) a1 += p[i];

    float tot;
#ifdef HAVE_WMMA_F32X4
    v2f A; A.x = a0; A.y = a1;          // A is 16x4 f32: 2 VGPRs/lane -> covers all 64 accumulators
    v2f Bones; Bones.x = 1.0f; Bones.y = 1.0f;
    v8f C = {};
    v8f D = __builtin_amdgcn_wmma_f32_16x16x4_f32(
        /*neg_a=*/false, A, /*neg_b=*/false, Bones,
        /*c_mod=*/(short)0, C, /*reuse_a=*/false, /*reuse_b=*/false);
    float t = D[0] + D[1] + D[2] + D[3] + D[4] + D[5] + D[6] + D[7];
    tot = t + __shfl_xor(t, 16, 32);
#else
    float t = a0 + a1;
    for (int off = 16; off > 0; off >>= 1)
        t += __shfl_xor(t, off, 32);
    tot = t;
#endif

    if (lane == 0) {
        const float vol = tot * (1.0f / 6.0f);
        invScale[b] = 1.0f / cbrtf(vol);
    }
}

// ---------------------------------------------------------------------------
// Kernel 3: streaming float4 scale: out = x * invScale[b]. 150000 floats per
// batch = exactly 37500 float4 -> pure b128 load/store traffic at HBM rate.
// ---------------------------------------------------------------------------
__global__ void __launch_bounds__(TPB) scale_kernel(
    const float* __restrict__ x, const float* __restrict__ invScale,
    float* __restrict__ out, int f4PerBatch, int perBatch)
{
    const int b = blockIdx.y;
    const float inv = invScale[b];
    const int i = blockIdx.x * TPB + threadIdx.x;
    if (i < f4PerBatch) {
        const float4* __restrict__ xi =
            (const float4*)(x + (size_t)b * perBatch);
        float4* __restrict__ oi = (float4*)(out + (size_t)b * perBatch);
        float4 v = xi[i];
        v.x *= inv; v.y *= inv; v.z *= inv; v.w *= inv;
        oi[i] = v;
    }
}

extern "C" void kernel_launch(void* const* d_in, const int* in_sizes, int n_in,
                              void* d_out, int out_size, void* d_ws, size_t ws_size,
                              hipStream_t stream)
{
    const float* x = (const float*)d_in[0];          // (64, 150000) f32
    const long long* M = (const long long*)d_in[1];  // (100000, 3) int64
    float* out = (float*)d_out;                      // (64, 150000) f32

    const int nTri = in_sizes[1] / 3;                 // 100000
    const int perBatch = in_sizes[0] / NBATCH;        // 150000
    const int nChunks = (nTri + TRI_PER_BLOCK - 1) / TRI_PER_BLOCK;  // 98

    float* partials = (float*)d_ws;                          // 64*nChunks floats
    float* invScale = partials + (size_t)NBATCH * nChunks;   // 64 floats

    dim3 g1(nChunks, NBATCH);
    vol_partial_kernel<<<g1, TPB, 0, stream>>>(x, M, partials, nTri, nChunks, perBatch);

    vol_finalize_kernel<<<NBATCH, 32, 0, stream>>>(partials, invScale, nChunks);

    const int f4 = perBatch / 4;
    dim3 g3((f4 + TPB - 1) / TPB, NBATCH);
    scale_kernel<<<g3, TPB, 0, stream>>>(x, invScale, out, f4, perBatch);
}